// pRNN_45492293599807
// MI455X (gfx1250) — compile-verified
//
#include <hip/hip_runtime.h>

#define B_  1024
#define I_  256
#define N_  2048
#define D_  32
#define C_  32
#define ROWS 128   // batch rows per block (8 waves * 16 rows)

typedef __attribute__((ext_vector_type(2))) float v2f;
typedef __attribute__((ext_vector_type(8))) float v8f;

#if defined(__has_builtin)
#if __has_builtin(__builtin_amdgcn_wmma_f32_16x16x4_f32)
#define HAVE_WMMA_F32X4 1
#endif
#endif

__global__ __launch_bounds__(256)
void prnn_last_node_kernel(const float* __restrict__ x,       // (B, I)
                           const float* __restrict__ h0,      // (B, N, D)
                           const float* __restrict__ conv_w,  // (I)
                           const float* __restrict__ conv_b,  // (I)
                           const float* __restrict__ W,       // (N, D, C)
                           const float* __restrict__ bias,    // (N, D)
                           const float* __restrict__ W_out,   // (1, C)
                           const float* __restrict__ b_out,   // (1)
                           const int*   __restrict__ idx,     // (N, C)
                           float* __restrict__ out_nodes,     // (B, D)   = d_out[0 .. B*D)
                           float* __restrict__ out2)          // (B)      = d_out[B*D ..)
{
    // gathered rows for this block's batch tile, padded to kill bank conflicts
    __shared__ float gLDS[ROWS][C_ + 1];
    // Bt[c][d] = W[N-1][d][c]  (the GEMM B-matrix, K=c along rows)
    __shared__ float Bt[C_][D_];
    __shared__ float wout[C_];

    const int tid  = threadIdx.x;
    const int row0 = blockIdx.x * ROWS;

    // ---- stage W[N-1]^T and W_out into LDS ----
    const float* Wlast = W + (size_t)(N_ - 1) * D_ * C_;
    for (int i = tid; i < D_ * C_; i += blockDim.x) {
        const int d = i / C_;
        const int c = i % C_;
        Bt[c][d] = Wlast[i];
    }
    if (tid < C_) wout[tid] = W_out[tid];

    // ---- gather phase: gathered[b, c] for the last node (n = N-1) ----
    // j < I  -> relu(x[b,j]*conv_w[j]+conv_b[j])   (the "inp" half of buf)
    // j >= I -> h0 flat at (j - I)                 (the "h0" half of buf)
    for (int i = tid; i < ROWS * C_; i += blockDim.x) {
        const int r = i / C_;
        const int c = i % C_;
        const int b = row0 + r;
        const int j = idx[(N_ - 1) * C_ + c];
        float v;
        if (j < I_) {
            const float t = fmaf(x[(size_t)b * I_ + j], conv_w[j], conv_b[j]);
            v = t > 0.0f ? t : 0.0f;
        } else {
            v = h0[(size_t)b * (N_ * D_) + (j - I_)];
        }
        gLDS[r][c] = v;
    }
    __syncthreads();

    // ---- out2 = relu(g @ W_out^T + b_out): one thread per row ----
    if (tid < ROWS) {
        float acc = b_out[0];
        #pragma unroll
        for (int c = 0; c < C_; ++c) acc = fmaf(gLDS[tid][c], wout[c], acc);
        out2[row0 + tid] = acc > 0.0f ? acc : 0.0f;
    }

    // ---- GEMM: out[b, d] = relu( sum_c g[b,c] * W[N-1,d,c] + bias[N-1,d] ) ----
    const float* bl = bias + (size_t)(N_ - 1) * D_;

#if defined(HAVE_WMMA_F32X4)
    // Each wave computes a 16(row) x 32(d) tile with V_WMMA_F32_16X16X4_F32.
    const int wave  = tid >> 5;
    const int lane  = tid & 31;
    const int rbase = wave * 16;          // first batch row of this wave's tile
    const int m     = lane & 15;          // M / N index within 16
    const int khalf = lane >> 4;          // lanes 0-15: K pair {0,1}; lanes 16-31: {2,3}

    v8f acc0 = {};   // d = 0..15
    v8f acc1 = {};   // d = 16..31
    #pragma unroll
    for (int k = 0; k < C_; k += 4) {
        const int kk = k + khalf * 2;
        v2f a;                                    // A 16x4: lane m holds row m, K pair per khalf
        a.x = gLDS[rbase + m][kk + 0];
        a.y = gLDS[rbase + m][kk + 1];
        v2f b0, b1;                               // B 4x16: lane m holds col m, K pair per khalf
        b0.x = Bt[kk + 0][m];
        b0.y = Bt[kk + 1][m];
        b1.x = Bt[kk + 0][16 + m];
        b1.y = Bt[kk + 1][16 + m];
        acc0 = __builtin_amdgcn_wmma_f32_16x16x4_f32(false, a, false, b0,
                                                     (short)0, acc0, false, false);
        acc1 = __builtin_amdgcn_wmma_f32_16x16x4_f32(false, a, false, b1,
                                                     (short)0, acc1, false, false);
    }

    // C/D layout: VGPR v -> M = v (lanes 0-15) or v+8 (lanes 16-31), N = lane&15
    const int mOff = (lane < 16) ? 0 : 8;
    #pragma unroll
    for (int v = 0; v < 8; ++v) {
        const int gb = row0 + rbase + mOff + v;
        const float o0 = acc0[v] + bl[m];
        const float o1 = acc1[v] + bl[16 + m];
        out_nodes[(size_t)gb * D_ + m]      = o0 > 0.0f ? o0 : 0.0f;
        out_nodes[(size_t)gb * D_ + 16 + m] = o1 > 0.0f ? o1 : 0.0f;
    }
#else
    // Scalar fallback (keeps compile green if the f32 WMMA builtin is absent).
    for (int i = tid; i < ROWS * D_; i += blockDim.x) {
        const int r = i / D_;
        const int d = i % D_;
        float acc = bl[d];
        #pragma unroll
        for (int c = 0; c < C_; ++c) acc = fmaf(gLDS[r][c], Bt[c][d], acc);
        out_nodes[(size_t)(row0 + r) * D_ + d] = acc > 0.0f ? acc : 0.0f;
    }
#endif
}

extern "C" void kernel_launch(void* const* d_in, const int* in_sizes, int n_in,
                              void* d_out, int out_size, void* d_ws, size_t ws_size,
                              hipStream_t stream) {
    (void)in_sizes; (void)n_in; (void)d_ws; (void)ws_size; (void)out_size;
    const float* x      = (const float*)d_in[0];
    const float* h0     = (const float*)d_in[1];
    const float* conv_w = (const float*)d_in[2];
    const float* conv_b = (const float*)d_in[3];
    const float* W      = (const float*)d_in[4];
    const float* bias   = (const float*)d_in[5];
    const float* W_out  = (const float*)d_in[6];
    const float* b_out  = (const float*)d_in[7];
    const int*   idx    = (const int*)d_in[8];

    float* out_nodes = (float*)d_out;              // B*D floats
    float* out2      = (float*)d_out + (size_t)B_ * D_;  // B floats

    dim3 grid(B_ / ROWS);   // 8 blocks
    dim3 block(256);        // 8 wave32s
    prnn_last_node_kernel<<<grid, block, 0, stream>>>(
        x, h0, conv_w, conv_b, W, bias, W_out, b_out, idx, out_nodes, out2);
}